// DynamicVoxelEncoder_60378650247528
// MI455X (gfx1250) — compile-verified
//
#include <hip/hip_runtime.h>
#include <stdint.h>

// ---------------- problem constants (match reference) ----------------
constexpr int      kN    = 2000000;
constexpr int      kB    = 4;
constexpr int      kDZ   = 42;
constexpr int      kDY   = 1442;
constexpr int      kDX   = 1442;
constexpr uint32_t kDYDX = (uint32_t)kDY * (uint32_t)kDX;          // 2,079,364
constexpr uint32_t kK    = (uint32_t)kDZ * kDYDX;                  // 87,333,288
constexpr uint32_t kSENT = (uint32_t)kB * kK;                      // 349,333,152
static_assert(kSENT % 32u == 0, "bitmap word alignment");
constexpr uint32_t kW    = kSENT / 32u;                            // 10,916,661 words (43.7 MB)

constexpr int      kTile = 256;                                    // points per block
constexpr uint32_t kWordsPerThread = 16;
constexpr uint32_t kScanBlock      = 256;
constexpr uint32_t kWordsPerBlock  = kScanBlock * kWordsPerThread; // 4096
constexpr uint32_t kNB             = (kW + kWordsPerBlock - 1) / kWordsPerBlock; // 2666

typedef __attribute__((ext_vector_type(2))) float        v2f;
typedef __attribute__((ext_vector_type(8))) float        v8f;
typedef __attribute__((ext_vector_type(4))) unsigned int v4u;
typedef __attribute__((ext_vector_type(8))) int          v8i;
typedef __attribute__((ext_vector_type(4))) int          v4i;

// ---------------- generic zero ----------------
__global__ void k_zero_u32(uint32_t* __restrict__ p, uint32_t n) {
  uint32_t i = blockIdx.x * blockDim.x + threadIdx.x;
  uint32_t s = gridDim.x * blockDim.x;
  for (; i < n; i += s) p[i] = 0u;
}

// ---------------- async global->LDS staging (per-lane async copy path) ----------------
__device__ __forceinline__ void stage_tile_async(const float* __restrict__ pts,
                                                 int tileBase, float* smem) {
  const uint32_t tid     = threadIdx.x;
  const uint32_t ldsBase = (uint32_t)(uintptr_t)smem;   // low 32 bits = LDS byte addr
  const uint32_t maxIdx  = (uint32_t)kN * 5u - 1u;
#pragma unroll
  for (int t = 0; t < 5; ++t) {
    uint32_t slot = tid + (uint32_t)t * 256u;           // 0..1279
    uint32_t gidx = (uint32_t)tileBase * 5u + slot;     // dword index into points
    if (gidx > maxIdx) gidx = maxIdx;                   // tail clamp (dup, unused)
    uint32_t voff = gidx * 4u;
    uint32_t lds  = ldsBase + slot * 4u;
    asm volatile("global_load_async_to_lds_b32 %0, %1, %2"
                 :
                 : "v"(lds), "v"(voff), "s"(pts)
                 : "memory");
  }
  asm volatile("s_wait_asynccnt 0x0" ::: "memory");
  __syncthreads();
}

// ---------------- TDM global->LDS staging (Tensor Data Mover path) ----------------
// One 1-D descriptor per block: tile_dim0 = 1280 dwords, OOB tail reads return 0.
__device__ __forceinline__ void stage_tile_tdm(const float* __restrict__ pts,
                                               int tileBase, float* smem) {
  if (threadIdx.x < 32) {                                // wave 0 issues the DMA
    const uint64_t gaddr = (uint64_t)(uintptr_t)(pts + (size_t)tileBase * 5);
    const uint32_t lds   = (uint32_t)(uintptr_t)smem;
    const uint32_t dim0  = (uint32_t)kN * 5u;            // tensor extent (elements)

    v4u g0;
    g0.x = 1u;                                           // count=1, user descriptor
    g0.y = lds;                                          // lds_addr (bytes)
    g0.z = (uint32_t)(gaddr & 0xFFFFFFFFu);              // global_addr[31:0]
    g0.w = (uint32_t)((gaddr >> 32) & 0x01FFFFFFu)       // global_addr[56:32]
         | (2u << 30);                                   // type = 2 ("image")

    v8i g1;
    g1[0] = (int)(2u << 16);                             // data_size = 2 (4 bytes)
    g1[1] = (int)((dim0 & 0xFFFFu) << 16);               // tensor_dim0[15:0] @ [63:48]
    g1[2] = (int)(((dim0 >> 16) & 0xFFFFu)               // tensor_dim0[31:16] @ [79:64]
         | (1u << 16));                                  // tensor_dim1 = 1 @ [111:80]
    g1[3] = (int)(1280u << 16);                          // tile_dim0 = 1280 @ [127:112]
    g1[4] = 0;                                           // tile_dim1 = tile_dim2 = 0
    g1[5] = (int)dim0;                                   // tensor_dim0_stride low 32
    g1[6] = 0;                                           // stride hi / dim1_stride lo
    g1[7] = 0;
    v4i z4 = {0, 0, 0, 0};
    v8i z8 = {0, 0, 0, 0, 0, 0, 0, 0};
    // amdgpu-toolchain (clang-23) arity: (g0, g1, g2, g3, g4, cpol)
    __builtin_amdgcn_tensor_load_to_lds(g0, g1, z4, z4, z8, 0);
    __builtin_amdgcn_s_wait_tensorcnt(0);
  }
  __syncthreads();
}

// ---------------- 256-element block inclusive scan on the matrix pipe ----------------
// Values are integer-valued and < 2^24, so f32 WMMA math is bit-exact.
// Step 1: P = M x U   (row-wise inclusive prefix;  U[k][n] = k<=n)
// Step 2: Z = L x Q + P  (adds exclusive prefix of row sums; L[m][k]=k<m,
//                         Q[k][n] = rowsum[k] broadcast)
__device__ __forceinline__ float block_scan256_wmma(float v, float* sh, float* shP) {
  sh[threadIdx.x] = v;
  __syncthreads();
  if (threadIdx.x < 32) {                 // wave 0, EXEC all ones (WMMA requirement)
    const int L    = threadIdx.x;
    const int half = L >> 4;              // K-half of the wave
    const int r    = L & 15;              // row m (A) / column n (B,D)
    v8f P = {};
#pragma unroll
    for (int c = 0; c < 4; ++c) {
      const int k0 = 4 * c + 2 * half;    // A: lanes0-15 K={4c,4c+1}; lanes16-31 {4c+2,4c+3}
      v2f A, Bv;
      A.x  = sh[r * 16 + k0];
      A.y  = sh[r * 16 + k0 + 1];
      Bv.x = (k0     <= r) ? 1.0f : 0.0f; // U chunk
      Bv.y = (k0 + 1 <= r) ? 1.0f : 0.0f;
      P = __builtin_amdgcn_wmma_f32_16x16x4_f32(false, A, false, Bv,
                                                (short)0, P, false, false);
    }
    // spill P (D-layout: element (vgpr + 8*half, r)) so row sums are readable
#pragma unroll
    for (int q = 0; q < 8; ++q) shP[(q + 8 * half) * 16 + r] = P[q];
    v8f Z = P;
#pragma unroll
    for (int c = 0; c < 4; ++c) {
      const int k0 = 4 * c + 2 * half;
      v2f A, Bv;
      A.x  = (k0     < r) ? 1.0f : 0.0f;  // strictly-lower ones
      A.y  = (k0 + 1 < r) ? 1.0f : 0.0f;
      Bv.x = shP[k0 * 16 + 15];           // rowsum[k0]   (broadcast along n)
      Bv.y = shP[(k0 + 1) * 16 + 15];     // rowsum[k0+1]
      Z = __builtin_amdgcn_wmma_f32_16x16x4_f32(false, A, false, Bv,
                                                (short)0, Z, false, false);
    }
#pragma unroll
    for (int q = 0; q < 8; ++q) shP[(q + 8 * half) * 16 + r] = Z[q];
  }
  __syncthreads();
  return shP[threadIdx.x];                // thread t = m*16+n -> scan[t]
}

// ---------------- pass 1: mark presence bitmap ----------------
__global__ void __launch_bounds__(256)
k_mark(const float* __restrict__ pts, const float* __restrict__ pcr,
       const float* __restrict__ vsz, uint32_t* __restrict__ bitmap) {
  __shared__ float smem[kTile * 5];
  const int tileBase = blockIdx.x * kTile;
  stage_tile_async(pts, tileBase, smem);

  const int i = tileBase + threadIdx.x;
  if (i >= kN) return;
  const float b = smem[threadIdx.x * 5 + 0];
  const float x = smem[threadIdx.x * 5 + 1];
  const float y = smem[threadIdx.x * 5 + 2];
  const float z = smem[threadIdx.x * 5 + 3];

  const float lox = pcr[0], loy = pcr[1], loz = pcr[2];
  const float hix = pcr[3], hiy = pcr[4], hiz = pcr[5];
  const bool keep = (x >= lox) & (x <= hix) & (y >= loy) & (y <= hiy) &
                    (z >= loz) & (z <= hiz);
  if (!keep) return;

  const int cx = (int)((x - lox) / vsz[0]);
  const int cy = (int)((y - loy) / vsz[1]);
  const int cz = (int)((z - loz) / vsz[2]);
  const uint32_t bi  = (uint32_t)(int)b;
  const uint32_t key = ((bi * (uint32_t)kDZ + (uint32_t)cz) * (uint32_t)kDY +
                        (uint32_t)cy) * (uint32_t)kDX + (uint32_t)cx;
  atomicOr(&bitmap[key >> 5], 1u << (key & 31u));
}

// ---------------- pass 2a: per-block popcount sums (WMMA scan, take total) ----------------
__global__ void __launch_bounds__(256)
k_scan_partial(const uint32_t* __restrict__ bitmap, uint32_t* __restrict__ partial) {
  __shared__ float shA[kScanBlock];
  __shared__ float shB[kScanBlock];
  const uint32_t tid  = threadIdx.x;
  const uint32_t base = blockIdx.x * kWordsPerBlock + tid * kWordsPerThread;
  uint32_t s = 0;
#pragma unroll
  for (uint32_t t = 0; t < kWordsPerThread; ++t) {
    uint32_t w = base + t;
    if (w < kW) s += __popc(bitmap[w]);
  }
  (void)block_scan256_wmma((float)s, shA, shB);
  if (tid == 0) partial[blockIdx.x] = (uint32_t)shB[kScanBlock - 1];
}

// ---------------- pass 2b: exclusive scan of block sums (single block) ----------------
__global__ void __launch_bounds__(256)
k_scan_offsets(uint32_t* __restrict__ partial, uint32_t n) {
  __shared__ float shA[kScanBlock];
  __shared__ float shB[kScanBlock];
  const uint32_t tid = threadIdx.x;
  uint32_t running = 0;
  for (uint32_t base = 0; base < n; base += kScanBlock) {
    const uint32_t i = base + tid;
    const uint32_t v = (i < n) ? partial[i] : 0u;
    const float incl = block_scan256_wmma((float)v, shA, shB);
    if (i < n) partial[i] = (uint32_t)incl - v + running;
    running += (uint32_t)shB[kScanBlock - 1];   // chunk total (uniform read)
  }
}

// ---------------- pass 2c: per-word exclusive popcount prefix ----------------
__global__ void __launch_bounds__(256)
k_scan_words(const uint32_t* __restrict__ bitmap, const uint32_t* __restrict__ partial,
             uint32_t* __restrict__ wordPrefix) {
  __shared__ float shA[kScanBlock];
  __shared__ float shB[kScanBlock];
  const uint32_t tid  = threadIdx.x;
  const uint32_t base = blockIdx.x * kWordsPerBlock + tid * kWordsPerThread;
  uint32_t pc[kWordsPerThread];
  uint32_t s = 0;
#pragma unroll
  for (uint32_t t = 0; t < kWordsPerThread; ++t) {
    uint32_t w = base + t;
    uint32_t p = (w < kW) ? __popc(bitmap[w]) : 0u;
    pc[t] = p;
    s += p;
  }
  const float incl = block_scan256_wmma((float)s, shA, shB);
  uint32_t excl = (uint32_t)incl - s + partial[blockIdx.x];
#pragma unroll
  for (uint32_t t = 0; t < kWordsPerThread; ++t) {
    uint32_t w = base + t;
    if (w < kW) wordPrefix[w] = excl;
    excl += pc[t];
  }
}

// ---------------- pass 3: accumulate sums (TDM-staged) ----------------
__global__ void __launch_bounds__(256)
k_accum(const float* __restrict__ pts, const float* __restrict__ pcr,
        const float* __restrict__ vsz, const uint32_t* __restrict__ bitmap,
        const uint32_t* __restrict__ wordPrefix,
        float* __restrict__ voxSums, uint32_t* __restrict__ cnts) {
  __shared__ float smem[kTile * 5];
  const int tileBase = blockIdx.x * kTile;
  stage_tile_tdm(pts, tileBase, smem);

  const int i = tileBase + threadIdx.x;
  if (i >= kN) return;
  const float b = smem[threadIdx.x * 5 + 0];
  const float x = smem[threadIdx.x * 5 + 1];
  const float y = smem[threadIdx.x * 5 + 2];
  const float z = smem[threadIdx.x * 5 + 3];
  const float e = smem[threadIdx.x * 5 + 4];

  const float lox = pcr[0], loy = pcr[1], loz = pcr[2];
  const float hix = pcr[3], hiy = pcr[4], hiz = pcr[5];
  const bool keep = (x >= lox) & (x <= hix) & (y >= loy) & (y <= hiy) &
                    (z >= loz) & (z <= hiz);
  if (!keep) return;

  const int cx = (int)((x - lox) / vsz[0]);
  const int cy = (int)((y - loy) / vsz[1]);
  const int cz = (int)((z - loz) / vsz[2]);
  const uint32_t bi  = (uint32_t)(int)b;
  const uint32_t key = ((bi * (uint32_t)kDZ + (uint32_t)cz) * (uint32_t)kDY +
                        (uint32_t)cy) * (uint32_t)kDX + (uint32_t)cx;

  const uint32_t w    = key >> 5;
  const uint32_t bit  = key & 31u;
  const uint32_t mask = (bit == 0u) ? 0u : (0xFFFFFFFFu >> (32u - bit));
  const uint32_t rank = wordPrefix[w] + __popc(bitmap[w] & mask);

  atomicAdd(&voxSums[(size_t)rank * 4 + 0], x);
  atomicAdd(&voxSums[(size_t)rank * 4 + 1], y);
  atomicAdd(&voxSums[(size_t)rank * 4 + 2], z);
  atomicAdd(&voxSums[(size_t)rank * 4 + 3], e);
  atomicAdd(&cnts[rank], 1u);
}

// ---------------- pass 4: finalize (mean, coords decode, valid) ----------------
__global__ void k_finalize(const uint32_t* __restrict__ bitmap,
                           const uint32_t* __restrict__ wordPrefix,
                           const uint32_t* __restrict__ cnts,
                           float* __restrict__ vox, float* __restrict__ coors,
                           float* __restrict__ valid) {
  uint32_t w = blockIdx.x * blockDim.x + threadIdx.x;
  const uint32_t stride = gridDim.x * blockDim.x;
  for (; w < kW; w += stride) {
    uint32_t m = bitmap[w];
    uint32_t r = wordPrefix[w];
    while (m) {
      const uint32_t bit = (uint32_t)__ffs(m) - 1u;
      m &= (m - 1u);
      const uint32_t key = (w << 5) | bit;
      const uint32_t c   = cnts[r];
      const float inv    = 1.0f / (float)(c > 0u ? c : 1u);
      vox[(size_t)r * 4 + 0] *= inv;
      vox[(size_t)r * 4 + 1] *= inv;
      vox[(size_t)r * 4 + 2] *= inv;
      vox[(size_t)r * 4 + 3] *= inv;
      const uint32_t bb  = key / kK;
      const uint32_t rem = key % kK;
      const uint32_t zz  = rem / kDYDX;
      const uint32_t yy  = (rem / (uint32_t)kDX) % (uint32_t)kDY;
      const uint32_t xx  = rem % (uint32_t)kDX;
      coors[(size_t)r * 4 + 0] = (float)bb;
      coors[(size_t)r * 4 + 1] = (float)zz;
      coors[(size_t)r * 4 + 2] = (float)yy;
      coors[(size_t)r * 4 + 3] = (float)xx;
      valid[r] = 1.0f;
      ++r;
    }
  }
}

// ---------------- launch ----------------
extern "C" void kernel_launch(void* const* d_in, const int* in_sizes, int n_in,
                              void* d_out, int out_size, void* d_ws, size_t ws_size,
                              hipStream_t stream) {
  const float* pts = (const float*)d_in[0];   // N x 5
  const float* pcr = (const float*)d_in[1];   // 6
  const float* vsz = (const float*)d_in[2];   // 3
  (void)in_sizes; (void)n_in; (void)ws_size;

  // workspace layout: bitmap | wordPrefix | cnts | partial   (~95.3 MB)
  uint32_t* bitmap     = (uint32_t*)d_ws;
  uint32_t* wordPrefix = bitmap + kW;
  uint32_t* cnts       = wordPrefix + kW;
  uint32_t* partial    = cnts + kN;

  float* out   = (float*)d_out;
  float* vox   = out;                          // N x 4 (means)
  float* coors = out + (size_t)kN * 4;         // N x 4
  float* valid = out + (size_t)kN * 8;         // N

  const int ZB = 8192, ZT = 256;
  hipLaunchKernelGGL(k_zero_u32, dim3(ZB), dim3(ZT), 0, stream, bitmap, kW);
  hipLaunchKernelGGL(k_zero_u32, dim3(ZB), dim3(ZT), 0, stream, cnts, (uint32_t)kN);
  hipLaunchKernelGGL(k_zero_u32, dim3(ZB), dim3(ZT), 0, stream,
                     (uint32_t*)d_out, (uint32_t)out_size);

  const int nTiles = (kN + kTile - 1) / kTile;
  hipLaunchKernelGGL(k_mark, dim3(nTiles), dim3(kTile), 0, stream,
                     pts, pcr, vsz, bitmap);

  hipLaunchKernelGGL(k_scan_partial, dim3(kNB), dim3(kScanBlock), 0, stream,
                     bitmap, partial);
  hipLaunchKernelGGL(k_scan_offsets, dim3(1), dim3(kScanBlock), 0, stream,
                     partial, kNB);
  hipLaunchKernelGGL(k_scan_words, dim3(kNB), dim3(kScanBlock), 0, stream,
                     bitmap, partial, wordPrefix);

  hipLaunchKernelGGL(k_accum, dim3(nTiles), dim3(kTile), 0, stream,
                     pts, pcr, vsz, bitmap, wordPrefix, vox, cnts);

  hipLaunchKernelGGL(k_finalize, dim3(4096), dim3(256), 0, stream,
                     bitmap, wordPrefix, cnts, vox, coors, valid);
}